// DIFM_88734024335841
// MI455X (gfx1250) — compile-verified
//
#include <hip/hip_runtime.h>
#include <hip/hip_bf16.h>

// ---------------- problem constants ----------------
constexpr int B   = 2048;
constexpr int E   = 50;
constexpr int F   = 20;
constexpr int D   = 64;
constexpr int TSS = 16;
constexpr int HID = 256;
constexpr int DT  = D + TSS;      // 80
constexpr int F1  = F + 1;        // 21
constexpr int DIN = 3 * D + TSS;  // 208
constexpr int R1  = B * E;        // 102400 rows (event attention)
constexpr int R2  = B * F1;       // 43008 rows (feature attention)

typedef __attribute__((ext_vector_type(2))) float v2f;
typedef __attribute__((ext_vector_type(8))) float v8f;

// ---------------------------------------------------------------------------
// K1 (fused): single gather pass produces BOTH FM reductions.
//   fm1[b,e,:]  = FM over features (per event) ++ time_linear(time2vec)
//   fm2[b,f,:]  = FM over events (per feature slot, incl. time2vec slot)
// Block = 128 threads: thread d = tid&63, chunk = tid>>6 handles 25 events.
// Each embedding row is gathered exactly once (vs twice in the split version),
// halving the dominant ~1 GB L2 gather traffic. Next-event rows are prefetched
// (global_prefetch_b8) to overlap random-row L2 latency with FMA work.
// ---------------------------------------------------------------------------
__global__ void k_fm_fused(const int* __restrict__ ids, const float* __restrict__ vals,
                           const float* __restrict__ table, const float* __restrict__ tseq,
                           const float* __restrict__ w0, const float* __restrict__ b0,
                           const float* __restrict__ tw, const float* __restrict__ tb,
                           const float* __restrict__ tlw, const float* __restrict__ tlb,
                           float* __restrict__ fm1, float* __restrict__ fm2) {
  const int b     = blockIdx.x;
  const int d     = threadIdx.x & (D - 1);
  const int chunk = threadIdx.x >> 6;          // 0 or 1
  const int EC    = E / 2;                     // 25 events per chunk
  const int e0    = chunk * EC;

  __shared__ float lte[2][D];
  __shared__ float l2s[F1][D];
  __shared__ float l2ss[F1][D];

  float s2[F1], ss2[F1];
#pragma unroll
  for (int f = 0; f < F1; ++f) { s2[f] = 0.f; ss2[f] = 0.f; }

  const float tc0 = (d == 0) ? w0[0] : tw[d - 1];
  const float tc1 = (d == 0) ? b0[0] : tb[d - 1];

  // double-buffered ids/values for the prefetch pipeline
  int   nid[F];
  float nv[F];
#pragma unroll
  for (int f = 0; f < F; ++f) {
    nid[f] = ids[(b * E + e0) * F + f];
    nv[f]  = vals[(b * E + e0) * F + f];
  }

  for (int it = 0; it < EC; ++it) {
    const int e = e0 + it;
    int   cid[F];
    float cv[F];
#pragma unroll
    for (int f = 0; f < F; ++f) { cid[f] = nid[f]; cv[f] = nv[f]; }

    if (it + 1 < EC) {
#pragma unroll
      for (int f = 0; f < F; ++f) {
        nid[f] = ids[(b * E + e + 1) * F + f];
        nv[f]  = vals[(b * E + e + 1) * F + f];
      }
      // prefetch next event's embedding rows (each lane covers its own d slot)
#pragma unroll
      for (int f = 0; f < F; ++f)
        __builtin_prefetch(&table[(long)nid[f] * D + d], 0, 0);
    }

    float s1 = 0.f, ss1 = 0.f;
#pragma unroll
    for (int f = 0; f < F; ++f) {
      float x = table[(long)cid[f] * D + d] * cv[f];
      s1 += x; ss1 += x * x;        // FM over features (this event)
      s2[f] += x; ss2[f] += x * x;  // FM over events (this feature slot)
    }
    fm1[(long)(b * E + e) * DT + d] = 0.5f * (s1 * s1 - ss1);

    // time2vec for this event
    const float t  = tseq[b * E + e];
    const float z  = t * tc0 + tc1;
    const float te = (d == 0) ? z : __sinf(z);
    s2[F] += te; ss2[F] += te * te; // slot F of fm2

    lte[chunk][d] = te;
    __syncthreads();
    if (d < TSS) {
      float acc = tlb[d];
#pragma unroll 8
      for (int k = 0; k < D; ++k) acc += lte[chunk][k] * tlw[d * D + k];
      fm1[(long)(b * E + e) * DT + D + d] = acc;
    }
    __syncthreads();
  }

  // combine the two chunks' fm2 partials
  if (chunk == 1) {
#pragma unroll
    for (int f = 0; f < F1; ++f) { l2s[f][d] = s2[f]; l2ss[f][d] = ss2[f]; }
  }
  __syncthreads();
  if (chunk == 0) {
#pragma unroll
    for (int f = 0; f < F1; ++f) {
      float S  = s2[f] + l2s[f][d];
      float SS = ss2[f] + l2ss[f][d];
      fm2[(long)(b * F1 + f) * D + d] = 0.5f * (S * S - SS);
    }
  }
}

// ---------------------------------------------------------------------------
// K3/K6: WMMA attention prep. One wave per 16-row tile of X [R, C].
//   Q = X@Wq, K = X@Wk, V = X@Wv (all CxC). Emits V tile and per-row
//   score[r] = (sum_n Q[r,n]*K[r,n]) * scale.  Uses v_wmma_f32_16x16x4_f32.
// ---------------------------------------------------------------------------
template <int C>
__global__ void k_attn(const float* __restrict__ X,
                       const float* __restrict__ Wq, const float* __restrict__ Wk,
                       const float* __restrict__ Wv,
                       float* __restrict__ Vout, float* __restrict__ sc, float scale) {
  constexpr int NT = C / 16;
  constexpr int KT = C / 4;
  const int r0   = blockIdx.x * 16;
  const int lane = threadIdx.x;   // 0..31, single wave
  const int m    = lane & 15;
  const int kh   = lane >> 4;     // 0: K pair {0,1}; 1: K pair {2,3}

  v8f prod = {};                  // elementwise Q*K accumulation (per lane)
  __shared__ float red[32][8];

  for (int nt = 0; nt < NT; ++nt) {
    const int n0 = nt * 16;
    v8f qa = {}, ka = {}, va = {};
#pragma unroll
    for (int kt = 0; kt < KT; ++kt) {
      const int k0 = kt * 4;
      // A fragment: 16x4 f32. lanes 0-15 -> K={k0,k0+1}; lanes 16-31 -> K={k0+2,k0+3}
      v2f a;
      a.x = X[(long)(r0 + m) * C + k0 + 2 * kh];
      a.y = X[(long)(r0 + m) * C + k0 + 2 * kh + 1];
      // B fragments: 4x16 f32, N striped across lanes, K across VGPRs per half
      v2f bq, bk, bv;
      bq.x = Wq[(k0 + 2 * kh) * C + n0 + m];
      bq.y = Wq[(k0 + 2 * kh + 1) * C + n0 + m];
      bk.x = Wk[(k0 + 2 * kh) * C + n0 + m];
      bk.y = Wk[(k0 + 2 * kh + 1) * C + n0 + m];
      bv.x = Wv[(k0 + 2 * kh) * C + n0 + m];
      bv.y = Wv[(k0 + 2 * kh + 1) * C + n0 + m];
      qa = __builtin_amdgcn_wmma_f32_16x16x4_f32(false, a, false, bq, (short)0, qa, false, false);
      ka = __builtin_amdgcn_wmma_f32_16x16x4_f32(false, a, false, bk, (short)0, ka, false, false);
      va = __builtin_amdgcn_wmma_f32_16x16x4_f32(false, a, false, bv, (short)0, va, false, false);
    }
#pragma unroll
    for (int i = 0; i < 8; ++i) prod[i] += qa[i] * ka[i];
    // store V tile: lane, VGPR i -> row i + 8*kh, col m
#pragma unroll
    for (int i = 0; i < 8; ++i) {
      int row = i + 8 * kh;
      Vout[(long)(r0 + row) * C + n0 + m] = va[i];
    }
  }

  // cross-lane reduction of per-row partial scores via LDS
#pragma unroll
  for (int i = 0; i < 8; ++i) red[lane][i] = prod[i];
  __syncthreads();
  if (lane < 16) {
    int row   = lane;                 // 0..15
    int vi    = row & 7;
    int lbase = (row < 8) ? 0 : 16;
    float s = 0.f;
#pragma unroll
    for (int l = 0; l < 16; ++l) s += red[lbase + l][vi];
    sc[r0 + row] = s * scale;
  }
}

// ---------------------------------------------------------------------------
// K5: masked softmax over events + weighted-V sum -> deep_in[:, 0:DT]
// ---------------------------------------------------------------------------
__global__ void k_softmax_his(const float* __restrict__ sc, const float* __restrict__ Vv,
                              const int* __restrict__ seq_len, float* __restrict__ deep_in) {
  int b   = blockIdx.x;
  int tid = threadIdx.x;      // 128 threads
  __shared__ float p[E];
  int sl = seq_len[b];
  if (tid < E) {
    float s = sc[b * E + tid];
    if (tid >= sl) s -= 1e12f;
    p[tid] = s;
  }
  __syncthreads();
  if (tid == 0) {
    float mx = p[0];
    for (int e = 1; e < E; ++e) mx = fmaxf(mx, p[e]);
    float sum = 0.f;
    for (int e = 0; e < E; ++e) { p[e] = __expf(p[e] - mx); sum += p[e]; }
    float inv = 1.0f / sum;
    for (int e = 0; e < E; ++e) p[e] *= inv;
  }
  __syncthreads();
  if (tid < DT) {
    float acc = 0.f;
    for (int e = 0; e < E; ++e) acc += p[e] * Vv[(long)(b * E + e) * DT + tid];
    deep_in[(long)b * DIN + tid] = acc;
  }
}

// ---------------------------------------------------------------------------
// K7: softmax over feature slots (no mask) + weighted-V -> deep_in[:, DT:DT+D]
// ---------------------------------------------------------------------------
__global__ void k_softmax_feat(const float* __restrict__ sc, const float* __restrict__ Vf,
                               float* __restrict__ deep_in) {
  int b   = blockIdx.x;
  int tid = threadIdx.x;      // 64 threads
  __shared__ float p[F1];
  if (tid < F1) p[tid] = sc[b * F1 + tid];
  __syncthreads();
  if (tid == 0) {
    float mx = p[0];
    for (int f = 1; f < F1; ++f) mx = fmaxf(mx, p[f]);
    float sum = 0.f;
    for (int f = 0; f < F1; ++f) { p[f] = __expf(p[f] - mx); sum += p[f]; }
    float inv = 1.0f / sum;
    for (int f = 0; f < F1; ++f) p[f] *= inv;
  }
  __syncthreads();
  float acc = 0.f;
  for (int f = 0; f < F1; ++f) acc += p[f] * Vf[(long)(b * F1 + f) * D + tid];
  deep_in[(long)b * DIN + DT + tid] = acc;
}

// ---------------------------------------------------------------------------
// K8: last-event embedding -> deep_in[:, DT+D:], and wide/LR reduction -> lr[b]
// ---------------------------------------------------------------------------
__global__ void k_pred_lr(const int* __restrict__ ids, const float* __restrict__ vals,
                          const float* __restrict__ table, const int* __restrict__ seq_len,
                          const float* __restrict__ wide, const float* __restrict__ lr_bias,
                          float* __restrict__ deep_in, float* __restrict__ lr) {
  int b   = blockIdx.x;
  int tid = threadIdx.x;      // 128 threads
  int e   = seq_len[b] - 1;
  if (tid < D) {
    float acc = 0.f;
    for (int f = 0; f < F; ++f) {
      int idx = (b * E + e) * F + f;
      acc += table[(long)ids[idx] * D + tid] * vals[idx];
    }
    deep_in[(long)b * DIN + DT + D + tid] = acc;
  }
  __shared__ float red[128];
  float s = 0.f;
  for (int i = tid; i < E * F; i += 128) {
    int idx = b * E * F + i;
    s += wide[ids[idx]] * vals[idx];
  }
  red[tid] = s;
  __syncthreads();
  for (int st = 64; st > 0; st >>= 1) {
    if (tid < st) red[tid] += red[tid + st];
    __syncthreads();
  }
  if (tid == 0) lr[b] = red[0] + lr_bias[0];
}

// ---------------------------------------------------------------------------
// K10: fc1 WMMA GEMM  H = relu(deep_in @ fc1_w^T + b).  One wave per 16x16 tile.
// ---------------------------------------------------------------------------
__global__ void k_fc1(const float* __restrict__ X, const float* __restrict__ W,
                      const float* __restrict__ bias, float* __restrict__ H) {
  const int r0   = blockIdx.x * 16;
  const int n0   = blockIdx.y * 16;
  const int lane = threadIdx.x;
  const int m    = lane & 15;
  const int kh   = lane >> 4;
  v8f acc = {};
#pragma unroll 4
  for (int kt = 0; kt < DIN / 4; ++kt) {
    const int k0 = kt * 4;
    v2f a, bw;
    a.x  = X[(long)(r0 + m) * DIN + k0 + 2 * kh];
    a.y  = X[(long)(r0 + m) * DIN + k0 + 2 * kh + 1];
    // B[k, n] = fc1_w[n, k]  (row-major [HID, DIN])
    bw.x = W[(long)(n0 + m) * DIN + k0 + 2 * kh];
    bw.y = W[(long)(n0 + m) * DIN + k0 + 2 * kh + 1];
    acc = __builtin_amdgcn_wmma_f32_16x16x4_f32(false, a, false, bw, (short)0, acc, false, false);
  }
#pragma unroll
  for (int i = 0; i < 8; ++i) {
    int row  = i + 8 * kh;
    float v  = acc[i] + bias[n0 + m];
    H[(long)(r0 + row) * HID + n0 + m] = fmaxf(v, 0.0f);
  }
}

// ---------------------------------------------------------------------------
// K11: fc2 dot + bias + LR + sigmoid -> out[b]
// ---------------------------------------------------------------------------
__global__ void k_final(const float* __restrict__ H, const float* __restrict__ w2,
                        const float* __restrict__ b2, const float* __restrict__ lr,
                        float* __restrict__ out) {
  int b   = blockIdx.x;
  int tid = threadIdx.x;      // 64 threads
  float s = 0.f;
  for (int n = tid; n < HID; n += 64) s += H[(long)b * HID + n] * w2[n];
  __shared__ float red[64];
  red[tid] = s;
  __syncthreads();
  for (int st = 32; st > 0; st >>= 1) {
    if (tid < st) red[tid] += red[tid + st];
    __syncthreads();
  }
  if (tid == 0) {
    float z = red[0] + b2[0] + lr[b];
    out[b] = 1.0f / (1.0f + __expf(-z));
  }
}

// ---------------------------------------------------------------------------
extern "C" void kernel_launch(void* const* d_in, const int* in_sizes, int n_in,
                              void* d_out, int out_size, void* d_ws, size_t ws_size,
                              hipStream_t stream) {
  const int*   ids     = (const int*)d_in[0];
  const float* vals    = (const float*)d_in[1];
  const int*   seq_len = (const int*)d_in[2];
  const float* tseq    = (const float*)d_in[3];
  const float* table   = (const float*)d_in[4];
  const float* wide    = (const float*)d_in[5];
  const float* lr_bias = (const float*)d_in[6];
  const float* W1_1    = (const float*)d_in[7];
  const float* W1_2    = (const float*)d_in[8];
  const float* W1_3    = (const float*)d_in[9];
  const float* W2_1    = (const float*)d_in[10];
  const float* W2_2    = (const float*)d_in[11];
  const float* W2_3    = (const float*)d_in[12];
  const float* t2v_w0  = (const float*)d_in[13];
  const float* t2v_b0  = (const float*)d_in[14];
  const float* t2v_w   = (const float*)d_in[15];
  const float* t2v_b   = (const float*)d_in[16];
  const float* tlw     = (const float*)d_in[17];
  const float* tlb     = (const float*)d_in[18];
  const float* fc1_w   = (const float*)d_in[19];
  const float* fc1_b   = (const float*)d_in[20];
  const float* fc2_w   = (const float*)d_in[21];
  const float* fc2_b   = (const float*)d_in[22];
  float* out = (float*)d_out;

  // workspace carving (floats)
  float* ws      = (float*)d_ws;
  float* fm1     = ws;                       // R1*DT
  float* Vv2     = fm1 + (size_t)R1 * DT;    // R1*DT
  float* sc2     = Vv2 + (size_t)R1 * DT;    // R1
  float* fm2     = sc2 + R1;                 // R2*D
  float* Vf      = fm2 + (size_t)R2 * D;     // R2*D
  float* scf     = Vf + (size_t)R2 * D;      // R2
  float* deep_in = scf + R2;                 // B*DIN
  float* H       = deep_in + (size_t)B * DIN;// B*HID
  float* lr      = H + (size_t)B * HID;      // B

  k_fm_fused<<<B, 128, 0, stream>>>(ids, vals, table, tseq, t2v_w0, t2v_b0,
                                    t2v_w, t2v_b, tlw, tlb, fm1, fm2);

  const float scale80 = 1.0f / sqrtf((float)DT);
  const float scale64 = 1.0f / sqrtf((float)D);
  k_attn<DT><<<R1 / 16, 32, 0, stream>>>(fm1, W2_1, W2_2, W2_3, Vv2, sc2, scale80);
  k_attn<D><<<R2 / 16, 32, 0, stream>>>(fm2, W1_1, W1_2, W1_3, Vf, scf, scale64);

  k_softmax_his<<<B, 128, 0, stream>>>(sc2, Vv2, seq_len, deep_in);
  k_softmax_feat<<<B, 64, 0, stream>>>(scf, Vf, deep_in);
  k_pred_lr<<<B, 128, 0, stream>>>(ids, vals, table, seq_len, wide, lr_bias, deep_in, lr);

  dim3 g_fc1(B / 16, HID / 16);
  k_fc1<<<g_fc1, 32, 0, stream>>>(deep_in, fc1_w, fc1_b, H);
  k_final<<<B, 64, 0, stream>>>(H, fc2_w, fc2_b, lr, out);
}